// MyLoss_17334488007121
// MI455X (gfx1250) — compile-verified
//
#include <hip/hip_runtime.h>

typedef __attribute__((ext_vector_type(2))) float v2f;
typedef __attribute__((ext_vector_type(8))) float v8f;

#define BATCH 128
#define DIM   784
#define NCLS  10
#define CPAD  16
#define LAMBDA 0.999f

__launch_bounds__(256, 1)
__global__ void ce_hesssum_kernel(const float* __restrict__ x,
                                  const float* __restrict__ Wg,
                                  const float* __restrict__ bg,
                                  const int*   __restrict__ yg,
                                  float* __restrict__ out)
{
    // LDS: W zero-padded to 16 classes (50176 B) + bias + labels + reduction pad
    __shared__ float ldsW[DIM * CPAD];
    __shared__ float ldsB[CPAD];
    __shared__ int   ldsY[BATCH];
    __shared__ float red[16];

    const int tid  = threadIdx.x;
    const int wave = tid >> 5;
    const int lane = tid & 31;

    // ---- Phase 0: stage W (zero-padded cols), b, y into LDS ----
    for (int i = tid; i < DIM * CPAD; i += 256) {
        const int k = i >> 4, n = i & 15;
        ldsW[i] = (n < NCLS) ? Wg[k * NCLS + n] : 0.0f;
    }
    if (tid < CPAD)  ldsB[tid] = (tid < NCLS) ? bg[tid] : 0.0f;
    if (tid < BATCH) ldsY[tid] = yg[tid];
    __syncthreads();

    // ---- Phase 1: logits tile = x[16w:16w+16, :] @ W  via V_WMMA_F32_16X16X4_F32
    // A operand layout (f32 16x4): lane = M row (mod 16); lanes 16-31 take K+2/K+3.
    const int  n    = lane & 15;            // also the M row for the A operand
    const int  klo  = (lane >> 4) << 1;     // 0 or 2
    const int  row0 = wave << 4;
    const float* xrow = x + (size_t)(row0 + n) * DIM + klo;

    v8f   acc = {};
    float srow_part = 0.0f;                 // partial row-sum of x (this lane's K slots)

#pragma unroll 4
    for (int k = 0; k < DIM; k += 4) {
        // A: contiguous aligned pair from x row (global_load_b64)
        const v2f a = *(const v2f*)(xrow + k);
        // B: W[k+klo][n], W[k+klo+1][n] from padded LDS
        v2f bm;
        bm.x = ldsW[(k + klo) * CPAD + n];
        bm.y = ldsW[(k + klo + 1) * CPAD + n];
        srow_part += a.x + a.y;
        acc = __builtin_amdgcn_wmma_f32_16x16x4_f32(
            /*neg_a=*/false, a, /*neg_b=*/false, bm,
            /*c_mod=*/(short)0, acc, /*reuse_a=*/false, /*reuse_b=*/false);
    }

    // Complete per-row sums s: lanes L and L^16 cover complementary K slots of row L&15
    const float s_full = srow_part + __shfl_xor(srow_part, 16, 32);

    // ---- Phase 2: per-row softmax / CE / closed-form Hessian-sum term ----
    // C layout: lane holds column n = lane&15; VGPR j holds row j (+8 for lanes 16-31).
    const bool  upper = (lane >= 16);
    const float bias  = ldsB[n];
    const bool  valid = (n < NCLS);

    float ce_acc = 0.0f, cv_acc = 0.0f;

#pragma unroll
    for (int j = 0; j < 8; ++j) {
        const float z    = acc[j] + bias;
        const int   mrow = j + (upper ? 8 : 0);
        const int   grow = row0 + mrow;

        // row max over the 16 lanes of this half-wave
        float zm = valid ? z : -__builtin_inff();
        zm = fmaxf(zm, __shfl_xor(zm, 8, 16));
        zm = fmaxf(zm, __shfl_xor(zm, 4, 16));
        zm = fmaxf(zm, __shfl_xor(zm, 2, 16));
        zm = fmaxf(zm, __shfl_xor(zm, 1, 16));

        const float e = valid ? expf(z - zm) : 0.0f;
        float Z = e;
        Z += __shfl_xor(Z, 8, 16);
        Z += __shfl_xor(Z, 4, 16);
        Z += __shfl_xor(Z, 2, 16);
        Z += __shfl_xor(Z, 1, 16);

        // q = sum_c p_c computed numerically (analytically 1 -> curvature ~ 0,
        // matching the reference's numerically-noisy Hessian sum)
        const float p = e / Z;
        float q = p;
        q += __shfl_xor(q, 8, 16);
        q += __shfl_xor(q, 4, 16);
        q += __shfl_xor(q, 2, 16);
        q += __shfl_xor(q, 1, 16);

        // cross-entropy: lane holding the label column contributes
        const int yr = ldsY[grow];
        if (n == yr) ce_acc += zm + logf(Z) - z;

        // 1^T H 1 row term: (s_i + 1)^2 * (q - q^2); one lane per row contributes
        if (n == 0) {
            const float s   = __shfl(s_full, mrow, 16);  // s of row (lane&15)==mrow
            const float sp1 = s + 1.0f;
            cv_acc += sp1 * sp1 * (q - q * q);
        }
    }

    // ---- Phase 3: reduce 256 threads -> 3 scalars ----
    for (int mask = 16; mask >= 1; mask >>= 1) {
        ce_acc += __shfl_xor(ce_acc, mask, 32);
        cv_acc += __shfl_xor(cv_acc, mask, 32);
    }
    if (lane == 0) { red[wave] = ce_acc; red[8 + wave] = cv_acc; }
    __syncthreads();

    if (tid == 0) {
        float ce = 0.0f, cv = 0.0f;
#pragma unroll
        for (int i = 0; i < 8; ++i) { ce += red[i]; cv += red[8 + i]; }
        ce *= (1.0f / BATCH);
        cv *= (1.0f / BATCH);
        out[0] = ce * LAMBDA + cv * (1.0f - LAMBDA);  // total
        out[1] = ce;                                  // cross-entropy
        out[2] = cv;                                  // curvature = sum(H)
    }
}

extern "C" void kernel_launch(void* const* d_in, const int* in_sizes, int n_in,
                              void* d_out, int out_size, void* d_ws, size_t ws_size,
                              hipStream_t stream) {
    (void)in_sizes; (void)n_in; (void)d_ws; (void)ws_size; (void)out_size;
    const float* x = (const float*)d_in[0];
    const float* W = (const float*)d_in[1];
    const float* b = (const float*)d_in[2];
    const int*   y = (const int*)d_in[3];
    float* out = (float*)d_out;
    ce_hesssum_kernel<<<1, 256, 0, stream>>>(x, W, b, y, out);
}